// VolumeRenderer_78804059947199
// MI455X (gfx1250) — compile-verified
//
#include <hip/hip_runtime.h>

typedef __attribute__((ext_vector_type(2))) float v2f;
typedef __attribute__((ext_vector_type(8))) float v8f;

#define KS 256
#define FE_SCALER 7.0f

static __device__ __forceinline__ v8f wmma_f32_16x16x4(v2f a, v2f b, v8f c) {
  // D = A(16x4,f32) * B(4x16,f32) + C(16x16,f32)
  return __builtin_amdgcn_wmma_f32_16x16x4_f32(false, a, false, b, (short)0, c,
                                               false, false);
}

__global__ __launch_bounds__(256, 1)
void volrend_kernel(const float* __restrict__ ray_start,
                    const float* __restrict__ ray_dir,
                    const float* __restrict__ sampled_depth,
                    const float* __restrict__ sampled_dists,
                    const int*   __restrict__ sampled_idx,
                    const float* __restrict__ voxel_color,
                    const float* __restrict__ W1,      // [6][64] row-major
                    const float* __restrict__ b1,      // [64]
                    const float* __restrict__ w_sigma, // [64]
                    const float* __restrict__ b_sigma, // [1]
                    const float* __restrict__ W_tex,   // [64][3]
                    const float* __restrict__ b_tex,   // [3]
                    float* __restrict__ colors,        // [N][3]
                    float* __restrict__ depths,        // [N]
                    float* __restrict__ missed,        // [N]
                    float* __restrict__ probs)         // [N][K]
{
  // LDS (all WMMA staging traffic is wave-local; barriers only guard the scan)
  __shared__ float featLDS[256][9];     // 8 used cols (feat padded 6->8), +1 pad
  __shared__ float hbuf[8][16][65];     // per-wave h tile, 65 = bank-conflict pad
  __shared__ float outLDS[256][4];      // sigma, r, g, b per sample
  __shared__ float waveSum[8];
  __shared__ float waveInc[8];
  __shared__ float red[8][8];

  const int tid  = threadIdx.x;
  const int lane = tid & 31;
  const int wv   = tid >> 5;
  const int n    = blockIdx.x;

  const int nl    = lane & 15;            // N (or M) index inside a fragment
  const int khalf = (lane < 16) ? 0 : 2;  // K split across lane halves

  // ---- Phase 1: per-sample features -> LDS (feat padded 6 -> 8 with zeros)
  const float rs0 = ray_start[n * 3 + 0];
  const float rs1 = ray_start[n * 3 + 1];
  const float rs2 = ray_start[n * 3 + 2];
  const float rd0 = ray_dir[n * 3 + 0];
  const float rd1 = ray_dir[n * 3 + 1];
  const float rd2 = ray_dir[n * 3 + 2];
  const float dep = sampled_depth[(size_t)n * KS + tid];

  featLDS[tid][0] = rs0 + rd0 * dep;
  featLDS[tid][1] = rs1 + rd1 * dep;
  featLDS[tid][2] = rs2 + rd2 * dep;
  featLDS[tid][3] = rd0;
  featLDS[tid][4] = rd1;
  featLDS[tid][5] = rd2;
  featLDS[tid][6] = 0.0f;
  featLDS[tid][7] = 0.0f;

  // ---- Layer-1 B fragments: W1 (6x64) padded to 8x64, 2 k-chunks x 4 n-chunks
  v2f b1frag[2][4];
#pragma unroll
  for (int kc = 0; kc < 2; ++kc) {
#pragma unroll
    for (int j = 0; j < 4; ++j) {
      const int k0  = 4 * kc + khalf;
      const int col = 16 * j + nl;
      float bx = (k0 + 0 < 6) ? W1[(k0 + 0) * 64 + col] : 0.0f;
      float by = (k0 + 1 < 6) ? W1[(k0 + 1) * 64 + col] : 0.0f;
      v2f b; b.x = bx; b.y = by;
      b1frag[kc][j] = b;
    }
  }
  float b1v[4];
#pragma unroll
  for (int j = 0; j < 4; ++j) b1v[j] = b1[16 * j + nl];

  // ---- Layer-2 B fragments: W2 = [w_sigma | W_tex] as 64x4 (N padded to 16)
  v2f b2frag[16];
#pragma unroll
  for (int t = 0; t < 16; ++t) {
    const int k0 = 4 * t + khalf;
    float bx = 0.0f, by = 0.0f;
    if (nl < 4) {
      bx = (nl == 0) ? w_sigma[k0 + 0] : W_tex[(k0 + 0) * 3 + (nl - 1)];
      by = (nl == 0) ? w_sigma[k0 + 1] : W_tex[(k0 + 1) * 3 + (nl - 1)];
    }
    v2f b; b.x = bx; b.y = by;
    b2frag[t] = b;
  }

  // ---- Phase 2: per-wave WMMA MLP over two 16-sample groups
  const int rb = (lane < 16) ? 0 : 8;   // row base of C/D fragment half
#pragma unroll
  for (int gl = 0; gl < 2; ++gl) {
    const int s0 = 32 * wv + 16 * gl;   // group's first sample (wave-local rows)

    // Layer 1: h[16x64] = feat[16x8] * W1p[8x64]
    v8f acc[4] = {};
#pragma unroll
    for (int kc = 0; kc < 2; ++kc) {
      v2f a;
      a.x = featLDS[s0 + nl][4 * kc + khalf + 0];
      a.y = featLDS[s0 + nl][4 * kc + khalf + 1];
#pragma unroll
      for (int j = 0; j < 4; ++j)
        acc[j] = wmma_f32_16x16x4(a, b1frag[kc][j], acc[j]);
    }
    // bias + relu, spill h tile to LDS in A-fragment-friendly layout
#pragma unroll
    for (int j = 0; j < 4; ++j) {
#pragma unroll
      for (int r = 0; r < 8; ++r)
        hbuf[wv][rb + r][16 * j + nl] = fmaxf(acc[j][r] + b1v[j], 0.0f);
    }

    // Layer 2: out[16x4] = h[16x64] * W2[64x4]  (16 chained WMMAs)
    v8f acc2 = {};
#pragma unroll
    for (int t = 0; t < 16; ++t) {
      v2f a2;
      a2.x = hbuf[wv][nl][4 * t + khalf + 0];
      a2.y = hbuf[wv][nl][4 * t + khalf + 1];
      acc2 = wmma_f32_16x16x4(a2, b2frag[t], acc2);
    }
    // store only the 4 valid output columns
    if (nl < 4) {
#pragma unroll
      for (int r = 0; r < 8; ++r)
        outLDS[s0 + rb + r][nl] = acc2[r];
    }
  }
  __syncthreads();

  // ---- Phase 3: sigma/texture epilogue per sample
  const int id   = sampled_idx[(size_t)n * KS + tid];
  const bool msk = (id != -1);
  const int  is  = msk ? id : 0;

  const float sg = outLDS[tid][0] + b_sigma[0];
  float t0 = 0.0f, t1 = 0.0f, t2 = 0.0f;
  if (msk) {
    t0 = outLDS[tid][1] + b_tex[0] + voxel_color[(size_t)is * 3 + 0];
    t1 = outLDS[tid][2] + b_tex[1] + voxel_color[(size_t)is * 3 + 1];
    t2 = outLDS[tid][3] + b_tex[2] + voxel_color[(size_t)is * 3 + 2];
  }
  const float dist = sampled_dists[(size_t)n * KS + tid];
  const float fe   = msk ? fmaxf(sg, 0.0f) * dist * FE_SCALER : 0.0f;

  // ---- Phase 4: exclusive prefix sum of fe over K=256 (wave32 scan + combine)
  float inc = fe;
#pragma unroll
  for (int d = 1; d < 32; d <<= 1) {
    float o = __shfl_up(inc, d, 32);
    if (lane >= d) inc += o;
  }
  if (lane == 31) waveSum[wv] = inc;
  __syncthreads();
  if (tid < 8) {
    float s = waveSum[tid];
#pragma unroll
    for (int d = 1; d < 8; d <<= 1) {
      float o = __shfl_up(s, d, 32);
      if (tid >= d) s += o;
    }
    waveInc[tid] = s;  // inclusive over wave totals
  }
  __syncthreads();
  const float excl = (inc - fe) + ((wv > 0) ? waveInc[wv - 1] : 0.0f);
  const float p    = (1.0f - expf(-fe)) * expf(-excl);

  probs[(size_t)n * KS + tid] = p;

  // ---- Phase 5: block reductions for colors / depth / missed
  float q0 = p, q1 = dep * p, q2 = t0 * p, q3 = t1 * p, q4 = t2 * p;
#pragma unroll
  for (int off = 16; off; off >>= 1) {
    q0 += __shfl_down(q0, off, 32);
    q1 += __shfl_down(q1, off, 32);
    q2 += __shfl_down(q2, off, 32);
    q3 += __shfl_down(q3, off, 32);
    q4 += __shfl_down(q4, off, 32);
  }
  if (lane == 0) {
    red[wv][0] = q0; red[wv][1] = q1; red[wv][2] = q2;
    red[wv][3] = q3; red[wv][4] = q4;
  }
  __syncthreads();
  if (tid < 5) {
    float s = 0.0f;
#pragma unroll
    for (int w = 0; w < 8; ++w) s += red[w][tid];
    if (tid == 0)      missed[n] = 1.0f - s;
    else if (tid == 1) depths[n] = s;
    else               colors[n * 3 + (tid - 2)] = s;
  }
}

extern "C" void kernel_launch(void* const* d_in, const int* in_sizes, int n_in,
                              void* d_out, int out_size, void* d_ws, size_t ws_size,
                              hipStream_t stream) {
  const float* ray_start     = (const float*)d_in[0];
  const float* ray_dir       = (const float*)d_in[1];
  const float* sampled_depth = (const float*)d_in[2];
  const float* sampled_dists = (const float*)d_in[3];
  const int*   sampled_idx   = (const int*)d_in[4];
  const float* voxel_color   = (const float*)d_in[5];
  const float* W1            = (const float*)d_in[6];
  const float* b1            = (const float*)d_in[7];
  const float* w_sigma       = (const float*)d_in[8];
  const float* b_sigma       = (const float*)d_in[9];
  const float* W_tex         = (const float*)d_in[10];
  const float* b_tex         = (const float*)d_in[11];

  const int N = in_sizes[0] / 3;  // 4096

  float* out      = (float*)d_out;
  float* colorsO  = out;                    // [N][3]
  float* depthsO  = out + (size_t)N * 3;    // [N]
  float* missedO  = out + (size_t)N * 4;    // [N]
  float* probsO   = out + (size_t)N * 5;    // [N][K]

  hipLaunchKernelGGL(volrend_kernel, dim3(N), dim3(256), 0, stream,
                     ray_start, ray_dir, sampled_depth, sampled_dists,
                     sampled_idx, voxel_color, W1, b1, w_sigma, b_sigma,
                     W_tex, b_tex, colorsO, depthsO, missedO, probsO);
}